// Model_87436944212014
// MI455X (gfx1250) — compile-verified
//
#include <hip/hip_runtime.h>

typedef __attribute__((ext_vector_type(2))) float v2f;
typedef __attribute__((ext_vector_type(4))) float v4f;
typedef __attribute__((ext_vector_type(8))) float v8f;

#define N_STREAM 5
#define P_DIM    512
#define P_TOT    2560
#define BATCH    32
#define LAMDA    0.9999f
#define YITA     0.5f
#define KAPPA    0.8f
#define N_ITER   20
#define SLOPE    0.01f

__device__ __forceinline__ float f_p(float x) {
  x = fminf(fmaxf(x, -1.0f), 1.0f);
  return x >= 0.0f ? x : SLOPE * x;
}

// ---------------------------------------------------------------------------
// Kernel 1: M_out = LAMDA*M + YITA * mask * u v^T   (pure streaming, float4)
// ---------------------------------------------------------------------------
__global__ __launch_bounds__(256) void form_memory_kernel(
    const float* __restrict__ M, const float* __restrict__ p,
    const float* __restrict__ p_prev, float* __restrict__ Mout) {
  const size_t Q4 = P_TOT / 4;                         // 640 float4 per row
  size_t idx = (size_t)blockIdx.x * 256 + threadIdx.x; // flat float4 index
  size_t q4 = idx % Q4;
  size_t t  = idx / Q4;
  int pr = (int)(t % P_TOT);
  int b  = (int)(t / P_TOT);

  const float* prow_p  = p      + (size_t)b * P_TOT;
  const float* prow_pp = p_prev + (size_t)b * P_TOT;
  float u = prow_p[pr] + prow_pp[pr];                  // row factor (L0/L2 hit)
  int q = (int)q4 * 4;
  bool mask = (pr / P_DIM) >= (q / P_DIM);             // block lower-triangular

  const v4f* Min4 = (const v4f*)(M + ((size_t)b * P_TOT + pr) * P_TOT) + q4;
  v4f m4 = __builtin_nontemporal_load(Min4);           // streamed once
  v4f out = LAMDA * m4;
  if (mask) {
    v4f pv  = *(const v4f*)(prow_p  + q);              // 10KB/batch, cached
    v4f pv2 = *(const v4f*)(prow_pp + q);
    v4f v = pv - pv2;
    out = out + (YITA * u) * v;
  }
  v4f* o = (v4f*)(Mout + ((size_t)b * P_TOT + pr) * P_TOT) + q4;
  __builtin_nontemporal_store(out, o);
}

// ---------------------------------------------------------------------------
// Kernel 2: h0 = f_p(query)
// ---------------------------------------------------------------------------
__global__ __launch_bounds__(256) void init_h_kernel(
    const float* __restrict__ query, float* __restrict__ h) {
  int i = blockIdx.x * 256 + threadIdx.x;
  if (i < BATCH * P_TOT) h[i] = f_p(query[i]);
}

// ---------------------------------------------------------------------------
// Kernel 3: one attractor step  h_out = f_p(KAPPA*h + h @ M)
// Batched matvec via V_WMMA_F32_16X16X4_F32 with A rows broadcast from h.
//   A (16x4 f32): VGPR0 = h[p0+koff], VGPR1 = h[p0+koff+1], koff = 2*(lane>=16)
//   B (4x16 f32): VGPR j = M[p0+koff+j, q0 + lane%16]
//   D (16x16 f32): all rows identical; VGPR0 = y[q0 + lane%16]
// Each wave runs 2 independent accumulator chains (32 q columns).
// ---------------------------------------------------------------------------
__global__ __launch_bounds__(256) void attractor_iter_kernel(
    const float* __restrict__ M, const float* __restrict__ h_in,
    float* __restrict__ h_out) {
  __shared__ float sh[P_TOT];
  const int b    = blockIdx.x;
  const int tid  = threadIdx.x;
  const int lane = tid & 31;
  const int wave = tid >> 5;

  // stage h[b,:] (10 KB) in LDS; 2560/256 = 10 uniform iterations
  for (int i = tid; i < P_TOT; i += 256)
    sh[i] = h_in[(size_t)b * P_TOT + i];
  __syncthreads();

  const float* Mb   = M + (size_t)b * P_TOT * P_TOT;
  const int    n    = lane & 15;            // N index within tile
  const int    koff = (lane >> 4) << 1;     // 0 for lanes 0-15, 2 for 16-31
  const int    q0   = blockIdx.y * 256 + wave * 32;

  v8f acc0 = {};
  v8f acc1 = {};
#pragma unroll 4
  for (int p0 = 0; p0 < P_TOT; p0 += 4) {
    v2f a;
    a.x = sh[p0 + koff];
    a.y = sh[p0 + koff + 1];
    const float* base = Mb + (size_t)(p0 + koff) * P_TOT + n;
    v2f b0, b1;
    b0.x = __builtin_nontemporal_load(base + q0);
    b0.y = __builtin_nontemporal_load(base + q0 + P_TOT);
    b1.x = __builtin_nontemporal_load(base + q0 + 16);
    b1.y = __builtin_nontemporal_load(base + q0 + 16 + P_TOT);
    acc0 = __builtin_amdgcn_wmma_f32_16x16x4_f32(false, a, false, b0,
                                                 (short)0, acc0, false, false);
    acc1 = __builtin_amdgcn_wmma_f32_16x16x4_f32(false, a, false, b1,
                                                 (short)0, acc1, false, false);
  }

  if (lane < 16) {  // lanes 16-31 hold duplicate rows of D
    int q = q0 + lane;
    float r0 = f_p(KAPPA * sh[q]      + acc0[0]);
    float r1 = f_p(KAPPA * sh[q + 16] + acc1[0]);
    h_out[(size_t)b * P_TOT + q]      = r0;
    h_out[(size_t)b * P_TOT + q + 16] = r1;
  }
}

// ---------------------------------------------------------------------------
extern "C" void kernel_launch(void* const* d_in, const int* in_sizes, int n_in,
                              void* d_out, int out_size, void* d_ws, size_t ws_size,
                              hipStream_t stream) {
  const float* p      = (const float*)d_in[0];
  const float* p_prev = (const float*)d_in[1];
  const float* query  = (const float*)d_in[2];
  const float* M      = (const float*)d_in[3];

  float* Mout    = (float*)d_out;                           // [B,P,P]
  float* h_final = Mout + (size_t)BATCH * P_TOT * P_TOT;    // [B,P] tail
  float* h0 = (float*)d_ws;                                 // ping
  float* h1 = h0 + BATCH * P_TOT;                           // pong

  // 1) memory update (also the only reader of original M)
  size_t total4 = (size_t)BATCH * P_TOT * (P_TOT / 4);
  form_memory_kernel<<<dim3((unsigned)(total4 / 256)), 256, 0, stream>>>(
      M, p, p_prev, Mout);

  // 2) h0 = f_p(query)
  init_h_kernel<<<dim3((BATCH * P_TOT + 255) / 256), 256, 0, stream>>>(query, h0);

  // 3) 20 attractor steps; one launch per step (grid-wide dependence on h)
  float* bufs[2] = {h0, h1};
  const float* cur = h0;
  for (int it = 0; it < N_ITER; ++it) {
    float* outp = (it == N_ITER - 1) ? h_final : bufs[(it + 1) & 1];
    attractor_iter_kernel<<<dim3(BATCH, P_TOT / 256), 256, 0, stream>>>(
        Mout, cur, outp);
    cur = outp;
  }
}